// simpleGNN_14087492731595
// MI455X (gfx1250) — compile-verified
//
#include <hip/hip_runtime.h>
#include <hip/hip_bf16.h>

typedef float v2f __attribute__((ext_vector_type(2)));
typedef float v8f __attribute__((ext_vector_type(8)));

#define D 128  // feature dim (both layers' input width)

// ---------------------------------------------------------------------------
// deg[i] = 1.0  (self loop)
__global__ void k_deg_init(float* __restrict__ deg, int n) {
    int i = blockIdx.x * blockDim.x + threadIdx.x;
    if (i < n) deg[i] = 1.0f;
}

// deg[dst[e]] += 1
__global__ void k_deg_edges(float* __restrict__ deg, const long long* __restrict__ dst, int e_cnt) {
    int e = blockIdx.x * blockDim.x + threadIdx.x;
    if (e < e_cnt) atomicAdd(&deg[(int)dst[e]], 1.0f);
}

// deg -> rsqrt(deg) in place
__global__ void k_dinv(float* __restrict__ deg, int n) {
    int i = blockIdx.x * blockDim.x + threadIdx.x;
    if (i < n) deg[i] = rsqrtf(deg[i]);
}

// ---------------------------------------------------------------------------
// H = X @ W  via V_WMMA_F32_16X16X4_F32.
// One block = 256 threads = 8 waves; block handles a 16-row stripe of X.
// Wave w computes the 16x16 tile at columns [16w, 16w+16).
// W (128x128, 64KB) and the 16x128 X tile (padded) are staged in LDS.
__global__ void __launch_bounds__(256)
k_gemm_wmma(const float* __restrict__ X, const float* __restrict__ W,
            float* __restrict__ H) {
    __shared__ float sW[D * D];        // row-major: sW[k*128 + n]
    __shared__ float sX[16 * 132];     // padded rows (132) -> conflict-free

    const int tid  = threadIdx.x;
    const int m0   = blockIdx.x * 16;

    // cooperative stage of W1 into LDS
    for (int i = tid; i < D * D; i += 256) sW[i] = W[i];
    // cooperative stage of 16x128 X tile
    for (int i = tid; i < 16 * D; i += 256) {
        int r = i >> 7, c = i & (D - 1);
        sX[r * 132 + c] = X[(size_t)(m0 + r) * D + c];
    }
    __syncthreads();

    const int wave = tid >> 5;         // 0..7 -> column tile
    const int lane = tid & 31;
    const int half = lane >> 4;        // 0: lanes 0-15, 1: lanes 16-31
    const int l    = lane & 15;
    const int n0   = wave * 16;

    v8f acc = {};
    #pragma unroll
    for (int k = 0; k < D; k += 4) {
        // A fragment (16x4 f32): lanes 0-15 carry K=k,k+1; lanes 16-31 carry K=k+2,k+3
        v2f a, b;
        const int ka = k + 2 * half;
        a.x = sX[l * 132 + ka];
        a.y = sX[l * 132 + ka + 1];
        // B fragment (4x16 f32): row-striped across lanes, mirrored K split
        b.x = sW[ka * D + n0 + l];
        b.y = sW[(ka + 1) * D + n0 + l];
        acc = __builtin_amdgcn_wmma_f32_16x16x4_f32(
            /*neg_a=*/false, a, /*neg_b=*/false, b,
            /*c_mod=*/(short)0, acc, /*reuse_a=*/false, /*reuse_b=*/false);
    }

    // C/D layout: VGPR v, lanes 0-15 -> row v; lanes 16-31 -> row v+8; col = l
    #pragma unroll
    for (int v = 0; v < 8; ++v) {
        H[(size_t)(m0 + v + 8 * half) * D + n0 + l] = acc[v];
    }
}

// ---------------------------------------------------------------------------
// out1[i][f] = h[i][f] * dinv[i]^2   (self-loop term doubles as zero-init)
__global__ void k_agg_init(float* __restrict__ out1, const float* __restrict__ h,
                           const float* __restrict__ dinv) {
    size_t idx = (size_t)blockIdx.x * blockDim.x + threadIdx.x;
    int i = (int)(idx >> 7);
    float dv = dinv[i];
    out1[idx] = h[idx] * dv * dv;
}

// wave per edge: out1[dst] += h[src] * dinv[src]*dinv[dst]  (128 feats, 4/lane)
__global__ void __launch_bounds__(256)
k_agg_edges(float* __restrict__ out1, const float* __restrict__ h,
            const float* __restrict__ dinv,
            const long long* __restrict__ src, const long long* __restrict__ dst) {
    const int wave = threadIdx.x >> 5;
    const int lane = threadIdx.x & 31;
    const long long e = (long long)blockIdx.x * 8 + wave;
    const int s = (int)src[e];
    const int d = (int)dst[e];
    const float norm = dinv[s] * dinv[d];
    const float4 v = ((const float4*)(h + (size_t)s * D))[lane];
    float* o = out1 + (size_t)d * D + lane * 4;
    atomicAdd(o + 0, v.x * norm);
    atomicAdd(o + 1, v.y * norm);
    atomicAdd(o + 2, v.z * norm);
    atomicAdd(o + 3, v.w * norm);
}

// h1 = relu(out1 + b1)
__global__ void k_bias_relu(float* __restrict__ h1, const float* __restrict__ out1,
                            const float* __restrict__ b1) {
    size_t idx = (size_t)blockIdx.x * blockDim.x + threadIdx.x;
    float v = out1[idx] + b1[idx & (D - 1)];
    h1[idx] = v > 0.0f ? v : 0.0f;
}

// pre[i] = dot(h1[i,:], W2)   — wave per node, shuffle reduce (wave32)
__global__ void __launch_bounds__(256)
k_matvec(float* __restrict__ pre, const float* __restrict__ h1,
         const float* __restrict__ W2) {
    const int wave = threadIdx.x >> 5;
    const int lane = threadIdx.x & 31;
    const int i = blockIdx.x * 8 + wave;
    const float* row = h1 + (size_t)i * D;
    float s = 0.0f;
    #pragma unroll
    for (int k = 0; k < 4; ++k) s += row[lane + 32 * k] * W2[lane + 32 * k];
    #pragma unroll
    for (int m = 16; m >= 1; m >>= 1) s += __shfl_xor(s, m, 32);
    if (lane == 0) pre[i] = s;
}

// out[i] = b2 + pre[i] * dinv[i]^2
__global__ void k_out_init(float* __restrict__ out, const float* __restrict__ pre,
                           const float* __restrict__ dinv, const float* __restrict__ b2,
                           int n) {
    int i = blockIdx.x * blockDim.x + threadIdx.x;
    if (i < n) {
        float dv = dinv[i];
        out[i] = b2[0] + pre[i] * dv * dv;
    }
}

// out[dst] += pre[src] * dinv[src]*dinv[dst]
__global__ void k_out_edges(float* __restrict__ out, const float* __restrict__ pre,
                            const float* __restrict__ dinv,
                            const long long* __restrict__ src,
                            const long long* __restrict__ dst, int e_cnt) {
    int e = blockIdx.x * blockDim.x + threadIdx.x;
    if (e < e_cnt) {
        int s = (int)src[e];
        int d = (int)dst[e];
        atomicAdd(&out[d], pre[s] * dinv[s] * dinv[d]);
    }
}

// ---------------------------------------------------------------------------
extern "C" void kernel_launch(void* const* d_in, const int* in_sizes, int n_in,
                              void* d_out, int out_size, void* d_ws, size_t ws_size,
                              hipStream_t stream) {
    const float*     x   = (const float*)d_in[0];
    const long long* ei  = (const long long*)d_in[1];   // int64 edge_index [2, E]
    const float*     W1  = (const float*)d_in[2];
    const float*     b1  = (const float*)d_in[3];
    const float*     W2  = (const float*)d_in[4];
    const float*     b2  = (const float*)d_in[5];
    float*           out = (float*)d_out;

    const int N = in_sizes[0] / D;       // 50000
    const int E = in_sizes[1] / 2;       // 800000
    const long long* srcI = ei;          // edge_index[0]
    const long long* dstI = ei + E;      // edge_index[1]

    // workspace layout (floats): dinv[N] | h[N*128] | out1[N*128] | pre[N]
    float* ws   = (float*)d_ws;
    float* dinv = ws;
    float* h    = dinv + N;
    float* out1 = h + (size_t)N * D;
    float* pre  = out1 + (size_t)N * D;

    const int T = 256;
    const int nBlocks  = (N + T - 1) / T;
    const int eBlocks  = (E + T - 1) / T;
    const size_t nfeat = (size_t)N * D;

    // degree -> dinv
    k_deg_init <<<nBlocks, T, 0, stream>>>(dinv, N);
    k_deg_edges<<<eBlocks, T, 0, stream>>>(dinv, dstI, E);
    k_dinv     <<<nBlocks, T, 0, stream>>>(dinv, N);

    // layer 1: h = x @ W1 (WMMA), then normalized scatter-add, bias+relu
    k_gemm_wmma<<<N / 16, T, 0, stream>>>(x, W1, h);
    k_agg_init <<<(int)(nfeat / T), T, 0, stream>>>(out1, h, dinv);
    k_agg_edges<<<E / 8, T, 0, stream>>>(out1, h, dinv, srcI, dstI);
    k_bias_relu<<<(int)(nfeat / T), T, 0, stream>>>(h, out1, b1);  // h := relu(out1+b1)

    // layer 2: matvec + normalized scatter-add
    k_matvec   <<<N / 8, T, 0, stream>>>(pre, h, W2);
    k_out_init <<<nBlocks, T, 0, stream>>>(out, pre, dinv, b2, N);
    k_out_edges<<<eBlocks, T, 0, stream>>>(out, pre, dinv, srcI, dstI, E);
}